// CoreAttention_16922171146362
// MI455X (gfx1250) — compile-verified
//
#include <hip/hip_runtime.h>
#include <math.h>

// Problem constants (B=1, H=16, S=2048, D=128), fp32 causal attention.
#define H_  16
#define S_  2048
#define D_  128
#define BR  128        // query rows per workgroup (8 waves x 16 rows)
#define BC  32         // key columns per tile
#define NWAVE 8        // 256 threads / wave32

#define KPAD 132       // K tile LDS row stride (floats): mult of 4 (b128 align), bank-spread
#define VPAD 136       // V tile LDS row stride (floats): mult of 4; 2*136 % 64 == 16 -> halves on disjoint banks
#define PPAD 34        // per-wave P scratch row stride

typedef float v2f  __attribute__((ext_vector_type(2)));
typedef float v8f  __attribute__((ext_vector_type(8)));

// Issue this wave's slice (4 rows) of one 32x128 f32 tile as async global->LDS b128 copies.
// 4 chunks x 16B per lane per matrix = 8 async ops per lane (ASYNCcnt += 8).
__device__ __forceinline__ void issue_async_tile(const float* __restrict__ kg,
                                                 const float* __restrict__ vg,
                                                 float* kbuf, float* vbuf,
                                                 int wave, int lane) {
#pragma unroll
  for (int i = 0; i < 4; ++i) {
    const int idx = wave * 128 + i * 32 + lane;   // float4-chunk id, 0..1023
    const int row = idx >> 5;                     // key row 0..31
    const int c4  = idx & 31;                     // float4 column 0..31
    const float* gk = kg + row * D_ + c4 * 4;
    const float* gv = vg + row * D_ + c4 * 4;
    const unsigned ldk = (unsigned)(size_t)&kbuf[row * KPAD + c4 * 4];
    const unsigned ldv = (unsigned)(size_t)&vbuf[row * VPAD + c4 * 4];
    asm volatile("global_load_async_to_lds_b128 %0, %1, off"
                 :: "v"(ldk), "v"(gk) : "memory");
    asm volatile("global_load_async_to_lds_b128 %0, %1, off"
                 :: "v"(ldv), "v"(gv) : "memory");
  }
}

__launch_bounds__(256, 1)
__global__ void fa_causal_f32_wmma(const float* __restrict__ Q,
                                   const float* __restrict__ K,
                                   const float* __restrict__ V,
                                   float* __restrict__ O) {
  __shared__ float k_lds[2][BC * KPAD];           // double-buffered, [key][dim] padded
  __shared__ float v_lds[2][BC * VPAD];           // double-buffered, [key][dim] padded
  __shared__ float p_lds[NWAVE * 16 * PPAD];      // per-wave 16x32 P scratch

  const int tid   = threadIdx.x;
  const int wave  = tid >> 5;
  const int lane  = tid & 31;
  const int lhalf = lane >> 4;     // 0: lanes 0-15, 1: lanes 16-31
  const int l16   = lane & 15;

  const int h     = blockIdx.x / (S_ / BR);
  const int qtile = blockIdx.x % (S_ / BR);
  const int q0w   = qtile * BR + wave * 16;       // first query row of this wave's slab

  const float scale = 0.08838834764831845f;       // 1/sqrt(128)

  const float* Kh = K + (size_t)h * S_ * D_;
  const float* Vh = V + (size_t)h * S_ * D_;

  // ---------------- Q fragments: A-matrix layout for 16x16x4 f32 WMMA ----------------
  v2f qf[32];
  {
    const float* qrow = Q + ((size_t)(h * S_ + q0w + l16)) * D_;
#pragma unroll
    for (int s = 0; s < 32; ++s) {
      const int kb = 4 * s + 2 * lhalf;
      qf[s].x = qrow[kb + 0];
      qf[s].y = qrow[kb + 1];
    }
  }

  // ---------------- accumulators / softmax state ----------------
  v8f oacc[8];
#pragma unroll
  for (int i = 0; i < 8; ++i) oacc[i] = (v8f)0.0f;
  float mrow[8], lrow[8];
#pragma unroll
  for (int v = 0; v < 8; ++v) { mrow[v] = -3.0e38f; lrow[v] = 0.0f; }

  const int nkt = (qtile + 1) * (BR / BC);        // causal tile bound
  float* pw = p_lds + wave * 16 * PPAD;

  // Prologue: start tile 0 into buffer 0.
  issue_async_tile(Kh, Vh, k_lds[0], v_lds[0], wave, lane);

  for (int kt = 0; kt < nkt; ++kt) {
    const int cur = kt & 1;
    // All waves finished reading buf[cur^1] (previous tile) before we overwrite it.
    __syncthreads();
    if (kt + 1 < nkt) {
      issue_async_tile(Kh + (size_t)(kt + 1) * BC * D_,
                       Vh + (size_t)(kt + 1) * BC * D_,
                       k_lds[cur ^ 1], v_lds[cur ^ 1], wave, lane);
      // 16 outstanding; wait until only the 8 newest (next tile) remain -> tile kt landed.
      asm volatile("s_wait_asynccnt 0x8" ::: "memory");
    } else {
      asm volatile("s_wait_asynccnt 0x0" ::: "memory");
    }
    __syncthreads();   // every wave's slice of tile kt is visible

    const float* kc = k_lds[cur];
    const float* vc = v_lds[cur];

    // ------------- S = Q * K^T  (16 x 32), f32 WMMA 16x16x4 -------------
    v8f sacc[2];
    sacc[0] = (v8f)0.0f;
    sacc[1] = (v8f)0.0f;
#pragma unroll
    for (int s = 0; s < 32; ++s) {
      const int kb = 4 * s + 2 * lhalf;
#pragma unroll
      for (int t = 0; t < 2; ++t) {
        const v2f bk = *(const v2f*)&kc[(t * 16 + l16) * KPAD + kb];
        sacc[t] = __builtin_amdgcn_wmma_f32_16x16x4_f32(
            false, qf[s], false, bk, (short)0, sacc[t], false, false);
      }
    }

    // ------------- scale + causal mask + online softmax -------------
    float corr[8];
#pragma unroll
    for (int v = 0; v < 8; ++v) {
      const int qrow = q0w + v + 8 * lhalf;
      float best = -3.0e38f;
#pragma unroll
      for (int t = 0; t < 2; ++t) {
        const int kidx = kt * BC + t * 16 + l16;
        float sc = sacc[t][v] * scale;
        sc = (kidx > qrow) ? -10000.0f : sc;
        sacc[t][v] = sc;
        best = fmaxf(best, sc);
      }
#pragma unroll
      for (int m = 1; m < 16; m <<= 1)
        best = fmaxf(best, __shfl_xor(best, m, 32));
      const float mnew = fmaxf(mrow[v], best);
      corr[v] = __expf(mrow[v] - mnew);
      mrow[v] = mnew;
    }
#pragma unroll
    for (int v = 0; v < 8; ++v) {
      float rs = 0.0f;
#pragma unroll
      for (int t = 0; t < 2; ++t) {
        const float p = __expf(sacc[t][v] - mrow[v]);
        sacc[t][v] = p;
        rs += p;
      }
#pragma unroll
      for (int m = 1; m < 16; m <<= 1)
        rs += __shfl_xor(rs, m, 32);
      lrow[v] = lrow[v] * corr[v] + rs;
    }
#pragma unroll
    for (int dt = 0; dt < 8; ++dt)
#pragma unroll
      for (int v = 0; v < 8; ++v) oacc[dt][v] *= corr[v];

    // ------------- transpose P (C-layout -> A-layout) via per-wave LDS -------------
#pragma unroll
    for (int t = 0; t < 2; ++t)
#pragma unroll
      for (int v = 0; v < 8; ++v)
        pw[(v + 8 * lhalf) * PPAD + t * 16 + l16] = sacc[t][v];
    asm volatile("s_wait_dscnt 0" ::: "memory");  // same-wave DS in-order visibility

    v2f pf[8];
#pragma unroll
    for (int kk = 0; kk < 8; ++kk) {
      const int kb = 4 * kk + 2 * lhalf;
      pf[kk] = *(const v2f*)&pw[l16 * PPAD + kb];
    }

    // ------------- O += P * V  (16 x 128), f32 WMMA 16x16x4 -------------
#pragma unroll
    for (int kk = 0; kk < 8; ++kk) {
      const int kb = 4 * kk + 2 * lhalf;          // key indices kb..kb+1
#pragma unroll
      for (int dt = 0; dt < 8; ++dt) {
        // B-fragment from row-major V: lane holds dim column N, keys kb..kb+1
        v2f bv;
        bv.x = vc[(kb + 0) * VPAD + dt * 16 + l16];
        bv.y = vc[(kb + 1) * VPAD + dt * 16 + l16];
        oacc[dt] = __builtin_amdgcn_wmma_f32_16x16x4_f32(
            false, pf[kk], false, bv, (short)0, oacc[dt], false, false);
      }
    }
  }

  // ---------------- finalize: O / l, write out ----------------
  float invl[8];
#pragma unroll
  for (int v = 0; v < 8; ++v) invl[v] = 1.0f / lrow[v];

  float* obase = O + ((size_t)(h * S_ + q0w)) * D_;
#pragma unroll
  for (int dt = 0; dt < 8; ++dt)
#pragma unroll
    for (int v = 0; v < 8; ++v)
      obase[(v + 8 * lhalf) * D_ + dt * 16 + l16] = oacc[dt][v] * invl[v];
}

extern "C" void kernel_launch(void* const* d_in, const int* in_sizes, int n_in,
                              void* d_out, int out_size, void* d_ws, size_t ws_size,
                              hipStream_t stream) {
  (void)in_sizes; (void)n_in; (void)d_ws; (void)ws_size; (void)out_size;
  const float* Q = (const float*)d_in[0];
  const float* K = (const float*)d_in[1];
  const float* V = (const float*)d_in[2];
  // d_in[3] = attention_mask (unused by reference), d_in[4] = dropout p (0.0)
  float* O = (float*)d_out;

  dim3 grid(H_ * (S_ / BR));   // 16 heads x 16 query tiles = 256 workgroups
  dim3 block(256);             // 8 waves (wave32)
  fa_causal_f32_wmma<<<grid, block, 0, stream>>>(Q, K, V, O);
}